// CRFLayer_2336462209784
// MI455X (gfx1250) — compile-verified
//
#include <hip/hip_runtime.h>
#include <hip/hip_bf16.h>

typedef __attribute__((ext_vector_type(2))) float v2f;
typedef __attribute__((ext_vector_type(8))) float v8f;

constexpr int KS = 32;    // number of tags/states
constexpr int TT = 512;   // sequence length
constexpr int NB = 1024;  // batch

// ---------------------------------------------------------------------------
// Forward scan, scaled-exponential domain. One wave32 per 16 batch rows.
// State: q[j][b] = exp(alpha[j][b] - C_b), kept transposed in WMMA C/D layout:
//   lane = b + 16*H, tile tt, vgpr v  ->  state j = tt*16 + 8*H + v
// Step: s^T = expT^T @ q^T  (16x V_WMMA_F32_16X16X4_F32, two 4-deep chains per
// tile), then q' = s * (exp(em) * 2^-e),  C += e*ln2.
// B operands are built from ONE shfl_xor(16) per q register (16 DS ops) plus
// one cndmask per slot: broadcast-from-low-half == (H ? swap : own) and
// broadcast-from-high-half == (H ? own : swap).
// The exponent proxy e comes from the FIRST partial WMMA chain so its DS
// shfl_xor overlaps the remaining 12 WMMAs. Any shift is exact: it cancels in
// log Z = C + log(sum q). No v_log in the loop; all v_exp depend only on loads.
// ---------------------------------------------------------------------------
__global__ __launch_bounds__(32)
void crf_forward_wmma(const float* __restrict__ em,
                      const unsigned char* __restrict__ mask,
                      const float* __restrict__ trans,
                      float* __restrict__ logz) {
    const int lane = threadIdx.x & 31;
    const int b    = lane & 15;          // batch-in-tile (N index)
    const int H    = lane >> 4;          // half select
    const int bg   = blockIdx.x * 16 + b;

    // ---- Constant A operands: A[k][mt] = expT^T, rows j=mt*16+0..15, cols i=4k..4k+3.
    // f32 A layout (16x4): lanes 0-15 carry (K=0,K=1) in (v0,v1); lanes 16-31 carry (K=2,K=3).
    v2f A[8][2];
    #pragma unroll
    for (int k = 0; k < 8; ++k) {
        const int i0 = 4 * k + 2 * H;
        #pragma unroll
        for (int mt = 0; mt < 2; ++mt) {
            const int j = mt * 16 + b;
            v2f a;
            a.x = __expf(trans[i0 * KS + j]);
            a.y = __expf(trans[(i0 + 1) * KS + j]);
            A[k][mt] = a;
        }
    }

    // ---- init: C = max_j em[b,0,j];  q = exp(em0 - C)
    const float* embase = em + (size_t)bg * TT * KS;
    float e0i[8], e1i[8];
    {
        float4 a0 = *(const float4*)(embase + 8 * H + 0);
        float4 a1 = *(const float4*)(embase + 8 * H + 4);
        float4 a2 = *(const float4*)(embase + 16 + 8 * H + 0);
        float4 a3 = *(const float4*)(embase + 16 + 8 * H + 4);
        e0i[0]=a0.x; e0i[1]=a0.y; e0i[2]=a0.z; e0i[3]=a0.w;
        e0i[4]=a1.x; e0i[5]=a1.y; e0i[6]=a1.z; e0i[7]=a1.w;
        e1i[0]=a2.x; e1i[1]=a2.y; e1i[2]=a2.z; e1i[3]=a2.w;
        e1i[4]=a3.x; e1i[5]=a3.y; e1i[6]=a3.z; e1i[7]=a3.w;
    }
    float C = e0i[0];
    #pragma unroll
    for (int v = 0; v < 8; ++v) { C = fmaxf(C, e0i[v]); C = fmaxf(C, e1i[v]); }
    C = fmaxf(C, __shfl_xor(C, 16, 32));
    float q[2][8];
    #pragma unroll
    for (int v = 0; v < 8; ++v) {
        q[0][v] = __expf(e0i[v] - C);
        q[1][v] = __expf(e1i[v] - C);
    }

    const unsigned char* mp = mask + (size_t)bg * TT;
    const float* emt = embase + KS;          // row t = 1

    for (int t = 1; t < TT; ++t, emt += KS) {
        if (t + 16 < TT) {                   // uniform branch: prefetch ahead
            __builtin_prefetch(emt + 16 * KS, 0, 1);
        }
        // ---- emission factors exp(em[t,j]) : depend only on loads (off-path)
        float4 a0 = *(const float4*)(emt + 8 * H + 0);
        float4 a1 = *(const float4*)(emt + 8 * H + 4);
        float4 a2 = *(const float4*)(emt + 16 + 8 * H + 0);
        float4 a3 = *(const float4*)(emt + 16 + 8 * H + 4);
        float ex0[8] = {__expf(a0.x), __expf(a0.y), __expf(a0.z), __expf(a0.w),
                        __expf(a1.x), __expf(a1.y), __expf(a1.z), __expf(a1.w)};
        float ex1[8] = {__expf(a2.x), __expf(a2.y), __expf(a2.z), __expf(a2.w),
                        __expf(a3.x), __expf(a3.y), __expf(a3.z), __expf(a3.w)};
        const bool mk = (mp[t] != 0);

        // ---- one cross-half swap per q register (16 DS ops total)
        float sw[2][8];
        #pragma unroll
        for (int tt = 0; tt < 2; ++tt)
            #pragma unroll
            for (int v = 0; v < 8; ++v) sw[tt][v] = __shfl_xor(q[tt][v], 16, 32);

        // ---- assemble 8 B operands with one cndmask per slot
        v2f Bv[8];
        #pragma unroll
        for (int k = 0; k < 8; ++k) {
            const int  tile = k >> 2;            // rows 4k..4k+3 -> source q tile
            const bool hi   = (k & 2) != 0;      // rows live in upper lane-half
            const int  s0   = (4 * k) & 7;       // vgpr group 0 or 4
            v2f bv;                               // B layout mirrors A's K split
            if (!hi) {   // broadcast from low half: own for H=0, swap for H=1
                bv.x = H ? sw[tile][s0 + 2] : q[tile][s0 + 0];
                bv.y = H ? sw[tile][s0 + 3] : q[tile][s0 + 1];
            } else {     // broadcast from high half: swap for H=0, own for H=1
                bv.x = H ? q[tile][s0 + 2] : sw[tile][s0 + 0];
                bv.y = H ? q[tile][s0 + 3] : sw[tile][s0 + 1];
            }
            Bv[k] = bv;
        }

        // ---- first chain, then launch the exponent proxy while the rest run
        v8f a0a = {0.f,0.f,0.f,0.f,0.f,0.f,0.f,0.f};
        #pragma unroll
        for (int k = 0; k < 4; ++k)
            a0a = __builtin_amdgcn_wmma_f32_16x16x4_f32(false, A[k][0], false, Bv[k],
                                                        (short)0, a0a, false, false);
        // partial-sum magnitude proxy (all terms positive -> within ~1 bit of s[0]);
        // its DS shfl_xor overlaps the 12 remaining WMMAs below.
        float r = fmaxf(a0a[0], __shfl_xor(a0a[0], 16, 32));   // uniform over lane pair
        const unsigned ebits = (__float_as_uint(r) >> 23) & 0xFFu;
        const float scale = __uint_as_float((254u - ebits) << 23);   // 2^-e
        const float dC = (float)((int)ebits - 127) * 0.69314718f;

        v8f a1a = {0.f,0.f,0.f,0.f,0.f,0.f,0.f,0.f};
        v8f a0b = {0.f,0.f,0.f,0.f,0.f,0.f,0.f,0.f};
        v8f a1b = {0.f,0.f,0.f,0.f,0.f,0.f,0.f,0.f};
        #pragma unroll
        for (int k = 0; k < 4; ++k)
            a1a = __builtin_amdgcn_wmma_f32_16x16x4_f32(false, A[k][1], false, Bv[k],
                                                        (short)0, a1a, false, false);
        #pragma unroll
        for (int k = 4; k < 8; ++k) {
            a0b = __builtin_amdgcn_wmma_f32_16x16x4_f32(false, A[k][0], false, Bv[k],
                                                        (short)0, a0b, false, false);
            a1b = __builtin_amdgcn_wmma_f32_16x16x4_f32(false, A[k][1], false, Bv[k],
                                                        (short)0, a1b, false, false);
        }
        // fold scale into emission factors while the tail WMMAs drain (off-path)
        float ex0s[8], ex1s[8];
        #pragma unroll
        for (int v = 0; v < 8; ++v) { ex0s[v] = ex0[v] * scale; ex1s[v] = ex1[v] * scale; }

        v8f s0v = a0a + a0b;
        v8f s1v = a1a + a1b;

        C = mk ? (C + dC) : C;

        // ---- q' = s * (exp(em) * 2^-e) : single multiply on the recurrence tail
        #pragma unroll
        for (int v = 0; v < 8; ++v) {
            float n0 = s0v[v] * ex0s[v];
            float n1 = s1v[v] * ex1s[v];
            q[0][v] = mk ? n0 : q[0][v];
            q[1][v] = mk ? n1 : q[1][v];
        }
    }

    // ---- log_z[b] = C + log(sum_j q[j])
    float s = 0.f;
    #pragma unroll
    for (int v = 0; v < 8; ++v) s += q[0][v] + q[1][v];
    s += __shfl_xor(s, 16, 32);
    float lz = C + __logf(s);
    if (lane < 16) logz[bg] = lz;
}

// ---------------------------------------------------------------------------
// Gold-path score: em_score + tr_score per batch (one block per batch row).
// ---------------------------------------------------------------------------
__global__ __launch_bounds__(256)
void crf_score(const float* __restrict__ em,
               const int* __restrict__ tags,
               const unsigned char* __restrict__ mask,
               const float* __restrict__ trans,
               float* __restrict__ scores) {
    const int bb  = blockIdx.x;
    const int tid = threadIdx.x;
    const size_t base = (size_t)bb * TT;
    float s = 0.f;
    for (int t = tid; t < TT; t += 256) {
        const int   tg = tags[base + t];
        const float mk = mask[base + t] ? 1.f : 0.f;
        s += em[(base + t) * KS + tg] * mk;
        if (t > 0) {
            const int tp = tags[base + t - 1];
            s += trans[tp * KS + tg] * mk;
        }
    }
    #pragma unroll
    for (int off = 16; off > 0; off >>= 1) s += __shfl_down(s, off, 32);
    __shared__ float red[8];
    if ((tid & 31) == 0) red[tid >> 5] = s;
    __syncthreads();
    if (tid < 8) {
        float v = red[tid];
        v += __shfl_down(v, 4, 32);
        v += __shfl_down(v, 2, 32);
        v += __shfl_down(v, 1, 32);
        if (tid == 0) scores[bb] = v;
    }
}

// ---------------------------------------------------------------------------
// loss = mean_b(log_z[b] - score[b])
// ---------------------------------------------------------------------------
__global__ __launch_bounds__(256)
void crf_final(const float* __restrict__ logz,
               const float* __restrict__ scores,
               float* __restrict__ out) {
    const int tid = threadIdx.x;
    float s = 0.f;
    for (int i = tid; i < NB; i += 256) s += logz[i] - scores[i];
    #pragma unroll
    for (int off = 16; off > 0; off >>= 1) s += __shfl_down(s, off, 32);
    __shared__ float red[8];
    if ((tid & 31) == 0) red[tid >> 5] = s;
    __syncthreads();
    if (tid < 8) {
        float v = red[tid];
        v += __shfl_down(v, 4, 32);
        v += __shfl_down(v, 2, 32);
        v += __shfl_down(v, 1, 32);
        if (tid == 0) out[0] = v / (float)NB;
    }
}

extern "C" void kernel_launch(void* const* d_in, const int* in_sizes, int n_in,
                              void* d_out, int out_size, void* d_ws, size_t ws_size,
                              hipStream_t stream) {
    const float*         em    = (const float*)d_in[0];         // (B,T,K) f32
    const int*           tags  = (const int*)d_in[1];           // (B,T) i32
    const unsigned char* mask  = (const unsigned char*)d_in[2]; // (B,T) bool
    const float*         trans = (const float*)d_in[3];         // (K,K) f32
    float* out    = (float*)d_out;
    float* logz   = (float*)d_ws;        // NB floats
    float* scores = logz + NB;           // NB floats

    crf_forward_wmma<<<NB / 16, 32, 0, stream>>>(em, mask, trans, logz);
    crf_score<<<NB, 256, 0, stream>>>(em, tags, mask, trans, scores);
    crf_final<<<1, 256, 0, stream>>>(logz, scores, out);
}